// AdaptiveEmbedding_60138132078902
// MI455X (gfx1250) — compile-verified
//
#include <hip/hip_runtime.h>

typedef __attribute__((ext_vector_type(2))) float v2f;
typedef __attribute__((ext_vector_type(8))) float v8f;

#define NTOK   (8 * 4096)
#define EMBED  1024
#define PAD    1
#define CUT0   20000
#define CUT1   40000

// ---------------------------------------------------------------------------
// Kernel 1: partition tokens into per-cluster compact index lists.
// Block-aggregated atomics: LDS counters per block, 3 global atomics per block.
// ---------------------------------------------------------------------------
__global__ __launch_bounds__(256) void partition_kernel(
    const int* __restrict__ ids, int* __restrict__ cnt,
    int* __restrict__ l0, int* __restrict__ l1, int* __restrict__ l2) {
  __shared__ int lcnt[3];
  __shared__ int lbase[3];
  const int tid = threadIdx.x;
  if (tid < 3) lcnt[tid] = 0;
  __syncthreads();
  const int i = blockIdx.x * 256 + tid;
  int c = 0, ofs = 0;
  if (i < NTOK) {
    const int id = ids[i];
    c = (id >= CUT1) ? 2 : (id >= CUT0) ? 1 : 0;
    ofs = atomicAdd(&lcnt[c], 1);
  }
  __syncthreads();
  if (tid < 3) lbase[tid] = atomicAdd(&cnt[tid], lcnt[tid]);
  __syncthreads();
  if (i < NTOK) {
    int* lst = (c == 0) ? l0 : (c == 1) ? l1 : l2;
    lst[lbase[c] + ofs] = i;
  }
}

// ---------------------------------------------------------------------------
// Kernel 2: cluster-0 tokens = direct 4KB row gather (float4, fully coalesced).
// One block (256 threads) copies one 1024-float row per list entry.
// ---------------------------------------------------------------------------
__global__ __launch_bounds__(256) void copy0_kernel(
    const int* __restrict__ ids, const float* __restrict__ emb0,
    const int* __restrict__ cntp, const int* __restrict__ l0,
    float* __restrict__ out) {
  const int count = *cntp;
  const int tid = threadIdx.x;
  for (int e = blockIdx.x; e < count; e += gridDim.x) {
    const int token = l0[e];
    const int id = ids[token];              // id in [0, CUT0)
    float4 v = make_float4(0.f, 0.f, 0.f, 0.f);
    if (id != PAD) v = ((const float4*)(emb0 + (size_t)id * EMBED))[tid];
    ((float4*)(out + (size_t)token * EMBED))[tid] = v;
  }
}

// ---------------------------------------------------------------------------
// Kernel 3: batched projection GEMM for clusters 1/2 via V_WMMA_F32_16X16X4_F32.
//   Out[16,1024] = E[16,DIM] x proj^T[DIM,1024]   (B[k][n] = proj[n*DIM + k])
// A tile staged in LDS with padded stride LSTRIDE (260/68) so the WMMA
// A-fragment reads (16 lanes, same k, different rows) hit 16 distinct banks.
// 4 waves/block; each wave owns 16 of the 64 output column-tiles.
// Tail rows (beyond count) get a scratch "dump" offset so all epilogue stores
// are unconditional. Offsets (not pointers) go through LDS so the stores stay
// kernarg-derived -> global_store_b32 (flat_store would also burn DScnt).
// ---------------------------------------------------------------------------
template <int DIM, int LSTRIDE>
__global__ __launch_bounds__(128) void proj_gemm_kernel(
    const int* __restrict__ ids, const float* __restrict__ table,
    const float* __restrict__ proj, const int* __restrict__ cntp,
    const int* __restrict__ list, int base, float* __restrict__ out,
    long long dumpOff) {
  const int count = *cntp;
  const int tile = blockIdx.x;
  if (tile * 16 >= count) return;          // uniform early-exit, no barriers yet

  __shared__ float E[16 * LSTRIDE];
  __shared__ long long rowOff[16];         // per-row float offset from `out`
  __shared__ int rowLoc[16];               // local table row, or -1 => zero row

  const int tid = threadIdx.x;
  if (tid < 16) {
    const int entry = tile * 16 + tid;
    int loc = -1;
    long long off = dumpOff + (long long)tid * EMBED;  // scratch row for tail
    if (entry < count) {
      const int tok = list[entry];
      off = (long long)tok * EMBED;
      const int l = ids[tok] - base;       // always in [0, tablesize) for listed tokens
      if (l != PAD) loc = l;               // padding_idx row is zero
    }
    rowOff[tid] = off;
    rowLoc[tid] = loc;
  }
  __syncthreads();

  // Stage A tile [16 x DIM] into LDS (float4), zero-filling invalid/pad rows.
  constexpr int SEGS = DIM / 4;
  for (int idx = tid; idx < 16 * SEGS; idx += 128) {
    const int row = idx / SEGS;
    const int seg = idx - row * SEGS;
    const int loc = rowLoc[row];
    float4 v = make_float4(0.f, 0.f, 0.f, 0.f);
    if (loc >= 0) v = ((const float4*)(table + (size_t)loc * DIM))[seg];
    ((float4*)(&E[row * LSTRIDE]))[seg] = v;
  }
  __syncthreads();

  const int wave = tid >> 5;
  const int lane = tid & 31;
  const int m    = lane & 15;              // A row / B column / D column
  const int hi   = lane >> 4;              // half-wave select

  // Hoist the 8 destination rows (D layout: M = r + 8*hi) as global pointers
  // derived from the kernarg `out` -> address space = global.
  float* dst[8];
#pragma unroll
  for (int r = 0; r < 8; ++r) dst[r] = out + rowOff[r + 8 * hi] + m;

  const float* arow = &E[m * LSTRIDE + 2 * hi];
  for (int ct = wave; ct < EMBED / 16; ct += 4) {
    const int colbase = ct * 16;
    const float* brow = proj + (size_t)(colbase + m) * DIM + 2 * hi;
    v8f acc = {};
#pragma unroll 4
    for (int kb = 0; kb < DIM; kb += 4) {
      const v2f a = *(const v2f*)(arow + kb);        // A: K = kb+2*hi+{0,1}, row m
      const v2f b = *(const v2f*)(brow + kb);        // B: K = kb+2*hi+{0,1}, col m
      acc = __builtin_amdgcn_wmma_f32_16x16x4_f32(
          /*neg_a=*/false, a, /*neg_b=*/false, b,
          /*c_mod=*/(short)0, acc, /*reuse_a=*/false, /*reuse_b=*/false);
    }
#pragma unroll
    for (int r = 0; r < 8; ++r) dst[r][colbase] = acc[r];   // unconditional stores
  }
}

// ---------------------------------------------------------------------------
extern "C" void kernel_launch(void* const* d_in, const int* in_sizes, int n_in,
                              void* d_out, int out_size, void* d_ws, size_t ws_size,
                              hipStream_t stream) {
  const int*   ids   = (const int*)d_in[0];
  const float* emb0  = (const float*)d_in[1];
  const float* emb1  = (const float*)d_in[2];
  const float* emb2  = (const float*)d_in[3];
  const float* proj1 = (const float*)d_in[4];
  const float* proj2 = (const float*)d_in[5];
  float*       out   = (float*)d_out;

  // workspace: [0,16) counters; lists (32768 ints each) at offset 128;
  // 64KB dump area (16 rows x 1024 f32) after the lists.
  int*   cnt  = (int*)d_ws;
  int*   l0   = (int*)((char*)d_ws + 128);
  int*   l1   = l0 + NTOK;
  int*   l2   = l1 + NTOK;
  float* dump = (float*)(l2 + NTOK);
  const long long dumpOff = (long long)(dump - out);   // float-element offset

  hipMemsetAsync(d_ws, 0, 16, stream);

  partition_kernel<<<NTOK / 256, 256, 0, stream>>>(ids, cnt, l0, l1, l2);
  copy0_kernel<<<1024, 256, 0, stream>>>(ids, emb0, cnt + 0, l0, out);
  proj_gemm_kernel<256, 260><<<NTOK / 16, 128, 0, stream>>>(
      ids, emb1, proj1, cnt + 1, l1, CUT0, out, dumpOff);
  proj_gemm_kernel<64, 68><<<NTOK / 16, 128, 0, stream>>>(
      ids, emb2, proj2, cnt + 2, l2, CUT1, out, dumpOff);
}